// SparseBPNeuralNetwork_34565896798955
// MI455X (gfx1250) — compile-verified
//
#include <hip/hip_runtime.h>
#include <cstddef>
#include <cstdint>

// ---------------------------------------------------------------------------
// Sparse-BP decoder as dense WMMA GEMMs (gfx1250 / CDNA5, wave32).
// B=128 codewords, N=1024 variable nodes, E=3072 edges.
//
//   C[B,O] = X[B,I] @ M^T,  M stored [O,I] row-major (or [I,O] when TRANSB).
//
// Activations live in a fragment-native swizzled layout: element (r,k) at
// ((r>>4)*(I/4) + (k>>2))*64 + ((k&2)<<4) + ((r&15)<<1) + (k&1), so one
// wave's 16x4 f32 A-fragment is a single contiguous 256B line (lane l ->
// float2 at base + 2*l), enabling clause'd global_load_b64 with imm offsets.
//
// Mask tiles are DOUBLE-BUFFERED in LDS via GLOBAL_LOAD_ASYNC_TO_LDS_B128
// (ASYNCcnt): issue fill of chunk k+1, compute 32 WMMAs on chunk k,
// s_wait_asynccnt 0, one barrier. 8 waves share the tile; each wave computes
// NT=2 output tiles per A fragment (6 WMMAs per 3 A-loads in the fused
// cn-update kernel).
// ---------------------------------------------------------------------------

typedef __attribute__((ext_vector_type(2))) float v2f;
typedef __attribute__((ext_vector_type(8))) float v8f;

#define BATCH 128
#define NVAR  1024
#define NEDGE 3072
#define KC    64          // K-chunk staged in LDS
#define LDSS  68          // LDS row stride (floats): bank-conflict free, 16B mult.
#define NT    2           // output tiles per wave
#define TILEF (NT * 16 * LDSS)   // floats per LDS tile buffer
#define CLIPV 0.999999f

enum { EP_NONE = 0, EP_OUT = 1, EP_VN = 2 };

__device__ __forceinline__ size_t swz_idx(int r, int c, int I) {
    return ((size_t)(r >> 4) * (size_t)(I >> 2) + (size_t)(c >> 2)) * 64
         + (size_t)((((c & 2) << 4) | ((r & 15) << 1) | (c & 1)));
}

__device__ __forceinline__ void wait_async0() {
#if __has_builtin(__builtin_amdgcn_s_wait_asynccnt)
    __builtin_amdgcn_s_wait_asynccnt(0);
#else
    asm volatile("s_wait_asynccnt 0x0" ::: "memory");
#endif
}

// Async-copy one (16*NT) x KC f32 mask tile (rows tileN.., cols kk..) into LDS.
// 512 x b128 transfers, 2 per thread; LDS dest rows padded to LDSS floats.
__device__ __forceinline__ void async_fill_mask(const float* __restrict__ Mk,
                                                float* buf, int tileN, int I,
                                                int kk) {
#pragma unroll
    for (int j = 0; j < 2; ++j) {
        int e4  = (int)threadIdx.x + 256 * j;      // 0..511
        int n32 = e4 >> 4;                         // 0..31 tile row
        int kb  = (e4 & 15) << 2;                  // 0..60 col (float idx)
        const float* g = &Mk[(size_t)(tileN + n32) * I + kk + kb];
        unsigned l = (unsigned)(uintptr_t)
            &buf[(n32 >> 4) * (16 * LDSS) + (n32 & 15) * LDSS + kb];
        asm volatile("global_load_async_to_lds_b128 %0, %1, off"
                     :: "v"(l), "v"(g) : "memory");
    }
}

// ---- fused cn-update GEMM: 3 swizzled A-matrices share one LDS mask tile --
// s = la@M^T ; negcnt = ng@M^T ; cnt = nz@M^T
// hmsg = 2*arctanh(clip(sign*exp(s) if cnt>0 else 0))   (hmsg swizzled)
__global__ void __launch_bounds__(256)
gemm_cn3(const float* __restrict__ la, const float* __restrict__ ng,
         const float* __restrict__ nz, const float* __restrict__ Mk,
         int I, int O, float* __restrict__ hmsg) {
    __shared__ __attribute__((aligned(16))) float smem[2 * TILEF];

    const int lane  = threadIdx.x & 31;
    const int wave  = threadIdx.x >> 5;
    const int tileN = blockIdx.x * (16 * NT);
    const int koff  = (lane >> 4) << 1;          // 0 or 2

    v8f accS[NT] = {}; v8f accG[NT] = {}; v8f accZ[NT] = {};

    const size_t fragBase = (size_t)wave * ((size_t)I << 4) + (size_t)(lane << 1);
    const float* A_S = la + fragBase;
    const float* A_G = ng + fragBase;
    const float* A_Z = nz + fragBase;

    async_fill_mask(Mk, smem, tileN, I, 0);
    wait_async0();
    __syncthreads();

    for (int kk = 0; kk < I; kk += KC) {
        float* cur = smem + ((kk / KC) & 1) * TILEF;
        float* nxt = smem + (1 - ((kk / KC) & 1)) * TILEF;
        if (kk + KC < I) async_fill_mask(Mk, nxt, tileN, I, kk + KC);

#pragma unroll
        for (int k4 = 0; k4 < KC; k4 += 4) {
            const size_t qo = (size_t)((kk + k4) >> 2) * 64;
            v2f aS = *(const v2f*)&A_S[qo];
            v2f aG = *(const v2f*)&A_G[qo];
            v2f aZ = *(const v2f*)&A_Z[qo];
#pragma unroll
            for (int t = 0; t < NT; ++t) {
                v2f bf = *(const v2f*)&cur[t * (16 * LDSS) + (lane & 15) * LDSS + k4 + koff];
                accS[t] = __builtin_amdgcn_wmma_f32_16x16x4_f32(false, aS, false, bf,
                                                                (short)0, accS[t], false, false);
                accG[t] = __builtin_amdgcn_wmma_f32_16x16x4_f32(false, aG, false, bf,
                                                                (short)0, accG[t], false, false);
                accZ[t] = __builtin_amdgcn_wmma_f32_16x16x4_f32(false, aZ, false, bf,
                                                                (short)0, accZ[t], false, false);
            }
        }
        wait_async0();       // own next-chunk fill landed
        __syncthreads();     // all waves done reading cur + all fills visible
    }

    const int ncol = lane & 15;
    const int rb   = (lane >> 4) << 3;
#pragma unroll
    for (int t = 0; t < NT; ++t) {
#pragma unroll
        for (int v = 0; v < 8; ++v) {
            int r = wave * 16 + rb + v;
            int c = tileN + t * 16 + ncol;
            float sv = accS[t][v], gv = accG[t][v], zv = accZ[t][v];
            float parity = gv - 2.0f * floorf(0.5f * gv);   // negcnt % 2
            float sign   = 1.0f - 2.0f * parity;
            float hval   = (zv > 0.0f) ? sign * expf(sv) : 0.0f;
            hval = fminf(fmaxf(hval, -CLIPV), CLIPV);
            hmsg[swz_idx(r, c, O)] = logf((1.0f + hval) / (1.0f - hval)); // 2*atanh
        }
    }
}

// ---- generic single-A GEMM (A swizzled) with fused epilogue ---------------
template <int EPI, bool TRANSB>
__global__ void __launch_bounds__(256)
gemm16(const float* __restrict__ X, const float* __restrict__ Mk,
       int I, int O, const float* __restrict__ extra,
       float* __restrict__ out0, float* __restrict__ out1,
       float* __restrict__ out2) {
    __shared__ __attribute__((aligned(16))) float smem[2 * TILEF];

    const int lane  = threadIdx.x & 31;
    const int wave  = threadIdx.x >> 5;
    const int tileN = blockIdx.x * (16 * NT);
    const int koff  = (lane >> 4) << 1;

    v8f acc[NT] = {};
    const float* A = X + (size_t)wave * ((size_t)I << 4) + (size_t)(lane << 1);

    auto compute_chunk = [&](const float* cur, int kk) {
#pragma unroll
        for (int k4 = 0; k4 < KC; k4 += 4) {
            v2f af = *(const v2f*)&A[(size_t)((kk + k4) >> 2) * 64];
#pragma unroll
            for (int t = 0; t < NT; ++t) {
                v2f bf = *(const v2f*)&cur[t * (16 * LDSS) + (lane & 15) * LDSS + k4 + koff];
                acc[t] = __builtin_amdgcn_wmma_f32_16x16x4_f32(false, af, false, bf,
                                                               (short)0, acc[t], false, false);
            }
        }
    };

    if constexpr (TRANSB) {
        // bias GEMM only: M stored [I,O]; column reads -> synchronous staging
        for (int kk = 0; kk < I; kk += KC) {
#pragma unroll
            for (int j = 0; j < (16 * NT * KC) / 256; ++j) {
                int e = (int)threadIdx.x + 256 * j;
                int n32 = e & 31, k = e >> 5;
                smem[(n32 >> 4) * (16 * LDSS) + (n32 & 15) * LDSS + k] =
                    Mk[(size_t)(kk + k) * O + tileN + n32];
            }
            __syncthreads();
            compute_chunk(smem, kk);
            __syncthreads();
        }
    } else {
        async_fill_mask(Mk, smem, tileN, I, 0);
        wait_async0();
        __syncthreads();
        for (int kk = 0; kk < I; kk += KC) {
            float* cur = smem + ((kk / KC) & 1) * TILEF;
            float* nxt = smem + (1 - ((kk / KC) & 1)) * TILEF;
            if (kk + KC < I) async_fill_mask(Mk, nxt, tileN, I, kk + KC);
            compute_chunk(cur, kk);
            wait_async0();
            __syncthreads();
        }
    }

    const int ncol = lane & 15;
    const int rb   = (lane >> 4) << 3;
#pragma unroll
    for (int t = 0; t < NT; ++t) {
#pragma unroll
        for (int v = 0; v < 8; ++v) {
            int r = wave * 16 + rb + v;
            int c = tileN + t * 16 + ncol;
            size_t idx = (size_t)r * O + c;
            float p = acc[t][v];
            if (EPI == EP_NONE) {         // bias = llr @ bias_mat (row-major out)
                out0[idx] = p;
            } else if (EPI == EP_OUT) {   // sigmoid(h @ out_mask^T + llr) -> d_out
                float s = p + extra[idx];
                out0[idx] = 1.0f / (1.0f + expf(-s));
            } else {                      // tanh(0.5*(h @ cn2vn^T + bias)) -> swizzled
                float th = tanhf(0.5f * (p + extra[idx]));
                size_t sidx = swz_idx(r, c, O);
                out0[sidx] = (th != 0.0f) ? logf(fabsf(th)) : 0.0f; // log|h|
                out1[sidx] = (th < 0.0f) ? 1.0f : 0.0f;             // neg
                out2[sidx] = (th != 0.0f) ? 1.0f : 0.0f;            // nz
            }
        }
    }
}

// ---- initial prep: swizzle x, and h=tanh(0.5*x) -> (log|h|, neg, nz) ------
__global__ void __launch_bounds__(256)
k_prep(const float* __restrict__ x, float* __restrict__ xs,
       float* __restrict__ la, float* __restrict__ ng, float* __restrict__ nz) {
    int i = blockIdx.x * 256 + threadIdx.x;          // i < B*N
    int r = i >> 10, c = i & (NVAR - 1);
    float xv = x[i];
    size_t d = swz_idx(r, c, NVAR);
    xs[d] = xv;
    float t = tanhf(0.5f * xv);
    la[d] = (t != 0.0f) ? logf(fabsf(t)) : 0.0f;
    ng[d] = (t < 0.0f) ? 1.0f : 0.0f;
    nz[d] = (t != 0.0f) ? 1.0f : 0.0f;
}

extern "C" void kernel_launch(void* const* d_in, const int* in_sizes, int n_in,
                              void* d_out, int out_size, void* d_ws, size_t ws_size,
                              hipStream_t stream) {
    const float* x          = (const float*)d_in[0];   // [B,N]
    const float* first_mask = (const float*)d_in[1];   // [E,N]
    const float* cn2vn      = (const float*)d_in[2];   // [E,E]
    const float* vn2cn      = (const float*)d_in[3];   // [E,E]
    const float* out_mask   = (const float*)d_in[4];   // [N,E]
    const float* bias_mat   = (const float*)d_in[5];   // [N,E]
    float* out = (float*)d_out;                        // 5 x [B,N], order out5..out1

    float* ws   = (float*)d_ws;                        // ~7.9 MB f32 scratch
    float* bias = ws;                                  // [B,E] row-major
    float* hmsg = bias + (size_t)BATCH * NEDGE;        // [B,E] swizzled
    float* la   = hmsg + (size_t)BATCH * NEDGE;        // [B,E] swizzled
    float* ng   = la   + (size_t)BATCH * NEDGE;        // [B,E] swizzled
    float* nz   = ng   + (size_t)BATCH * NEDGE;        // [B,E] swizzled
    float* xs   = hmsg;  // swizzled x [B,N]; consumed by bias GEMM before
                         // hmsg is first written (stream-serialized)

    dim3 blk(256);
    const size_t BN = (size_t)BATCH * NVAR;

    // swizzle x + initial message decomposition
    k_prep<<<(BATCH * NVAR) / 256, blk, 0, stream>>>(x, xs, la, ng, nz);

    // bias = llr @ bias_mat  (bias_mat stored [N,E] => TRANSB path)
    gemm16<EP_NONE, true><<<NEDGE / (16 * NT), blk, 0, stream>>>(
        xs, bias_mat, NVAR, NEDGE, nullptr, bias, nullptr, nullptr);

    // fc0: cn-update with first_mask [E,N], I = N
    gemm_cn3<<<NEDGE / (16 * NT), blk, 0, stream>>>(la, ng, nz, first_mask,
                                                    NVAR, NEDGE, hmsg);

    // out1 = sigmoid(hmsg @ out_mask^T + llr) -> last tuple slot
    gemm16<EP_OUT, false><<<NVAR / (16 * NT), blk, 0, stream>>>(
        hmsg, out_mask, NEDGE, NVAR, x, out + 4 * BN, nullptr, nullptr);

    for (int it = 0; it < 4; ++it) {
        // vn-update: tanh(0.5*(hmsg @ cn2vn^T + bias)) -> (la, ng, nz) swizzled
        gemm16<EP_VN, false><<<NEDGE / (16 * NT), blk, 0, stream>>>(
            hmsg, cn2vn, NEDGE, NEDGE, bias, la, ng, nz);
        // cn-update with vn2cn [E,E]
        gemm_cn3<<<NEDGE / (16 * NT), blk, 0, stream>>>(la, ng, nz, vn2cn,
                                                        NEDGE, NEDGE, hmsg);
        // out{2..5}: tuple (out5,out4,out3,out2,out1) -> offset (3-it)*BN
        gemm16<EP_OUT, false><<<NVAR / (16 * NT), blk, 0, stream>>>(
            hmsg, out_mask, NEDGE, NVAR, x, out + (size_t)(3 - it) * BN,
            nullptr, nullptr);
    }
}